// SlaterPooling_28776280883250
// MI455X (gfx1250) — compile-verified
//
#include <hip/hip_runtime.h>

typedef float v2f __attribute__((ext_vector_type(2)));
typedef float v8f __attribute__((ext_vector_type(8)));
typedef int   v4i __attribute__((ext_vector_type(4)));

#define NCFG 128
#define EDIM 16
#define MDIM 64
#define LDSTR 68  // 64 + 4 pad: keeps 16B row alignment, spreads banks

#if defined(__gfx1250__) && __has_builtin(__builtin_amdgcn_global_load_async_to_lds_b128)
#define HAVE_ASYNC_LDS 1
#else
#define HAVE_ASYNC_LDS 0
#endif

#if HAVE_ASYNC_LDS
typedef __attribute__((address_space(1))) v4i* gas_ptr_t;
typedef __attribute__((address_space(3))) v4i* las_ptr_t;
#endif

// fast hardware reciprocal (v_rcp_f32): ~1 ulp, fine for LU multipliers
__device__ __forceinline__ float fast_rcp(float x) {
#if defined(__gfx1250__) && __has_builtin(__builtin_amdgcn_rcpf)
  return __builtin_amdgcn_rcpf(x);
#else
  return 1.0f / x;
#endif
}

__global__ __launch_bounds__(256) void slater_det_kernel(
    const float* __restrict__ inp,   // (B, 64, 64) f32
    const int*   __restrict__ cfg,   // (128, 16) int32
    float*       __restrict__ out)   // (B, 128) f32
{
  __shared__ float As[MDIM * LDSTR];
  __shared__ int   cfgS[NCFG * EDIM];

  const int b    = blockIdx.x;
  const int tid  = threadIdx.x;
  const int lane = tid & 31;
  const int wave = tid >> 5;
  const int lh   = lane >> 4;   // half: rows 0-7 vs 8-15 of C layout
  const int col  = lane & 15;   // column owned by this lane

  // ---- stage configs into LDS ----
  for (int i = tid; i < NCFG * EDIM; i += 256) cfgS[i] = cfg[i];

  // ---- stage this batch's 64x64 matrix into LDS (async DMA path) ----
  {
    const int row = tid >> 2;
    const int q   = (tid & 3) * 16;
    const float* gsrc = inp + ((size_t)b * MDIM * MDIM) + row * MDIM + q;
    float* ldst = &As[row * LDSTR + q];
#if HAVE_ASYNC_LDS
    __builtin_amdgcn_global_load_async_to_lds_b128(
        (gas_ptr_t)gsrc, (las_ptr_t)ldst, 0, 0);
#if __has_builtin(__builtin_amdgcn_s_wait_asynccnt)
    __builtin_amdgcn_s_wait_asynccnt(0);
#else
    asm volatile("s_wait_asynccnt 0x0" ::: "memory");
#endif
#else
    const float4 v = *(const float4*)gsrc;
    *(float4*)ldst = v;
#endif
  }
  __syncthreads();

  // ---- each wave: 16 configs ----
  for (int c = wave; c < NCFG; c += 8) {
    const int* cf = &cfgS[c * EDIM];
    const int colidx = cf[col];
    // gather 16x16 submatrix directly in WMMA f32 C layout:
    // a[r] = A[cfg[r + 8*lh]][cfg[col]]
    float a[8];
#pragma unroll
    for (int r = 0; r < 8; ++r) {
      const int rowidx = cf[r + 8 * lh];
      a[r] = As[rowidx * LDSTR + colidx];
    }

    float det = 1.0f;

#pragma unroll
    for (int k0 = 0; k0 < 16; k0 += 4) {
      // ---- panel factorization: columns [k0, k0+4) ----
#pragma unroll
      for (int kk = 0; kk < 4; ++kk) {
        const int k  = k0 + kk;
        const int kr = k & 7;                 // VGPR holding row k
        const int ks = (k >= 8) ? 16 : 0;     // lane-half shift of row k
        const float piv = __shfl(a[kr], k + ks, 32);
        det *= piv;
        const float rpiv = fast_rcp(piv);
        // scale column k below the diagonal (L multipliers, stored in place)
#pragma unroll
        for (int r = 0; r < 8; ++r)
          if (col == k && (r + 8 * lh) > k) a[r] *= rpiv;
        // broadcast pivot row (U) to own column, and scaled column (L) per row
        const float urow = __shfl(a[kr], col + ks, 32);
#pragma unroll
        for (int r = 0; r < 8; ++r) {
          const float lc = __shfl(a[r], k + 16 * lh, 32);
          if ((r + 8 * lh) > k && col > k && col < k0 + 4)
            a[r] -= lc * urow;  // rank-1 update restricted to panel columns
        }
      }
      // ---- U12 solve: rows [k0,k0+4), cols >= k0+4 (forward subst.) ----
#pragma unroll
      for (int m = 0; m < 3; ++m) {
        const int km = k0 + m;
        const float urow = __shfl(a[km & 7], col + ((km >= 8) ? 16 : 0), 32);
#pragma unroll
        for (int x = m + 1; x < 4; ++x) {
          const int kx = k0 + x;
          const float lv = __shfl(a[kx & 7], km + ((kx >= 8) ? 16 : 0), 32);
          if (col >= k0 + 4 && lh == ((kx >= 8) ? 1 : 0))
            a[kx & 7] -= lv * urow;
        }
      }
      // ---- trailing update A22 -= L21*U12 via V_WMMA_F32_16X16X4_F32 ----
      if (k0 < 12) {
        const int kb = k0 + 4;
        // A-op (16x4): lane M = col, K = k0 + 2*lh + {0,1}; rows < kb zeroed
        float ax = 0.f, ay = 0.f;
#pragma unroll
        for (int t = 0; t < 2; ++t) {
          const int sl = (k0 + 2 * lh + t) + ((col & 8) ? 16 : 0);
#pragma unroll
          for (int r = 0; r < 8; ++r) {
            const float v = __shfl(a[r], sl, 32);
            if ((col & 7) == r) { if (t == 0) ax = v; else ay = v; }
          }
        }
        if (col < kb) { ax = 0.f; ay = 0.f; }
        // B-op (4x16): VGPR t, half lh -> row K = k0 + t + 2*lh; cols < kb zeroed
        float bx = 0.f, by = 0.f;
#pragma unroll
        for (int t = 0; t < 2; ++t) {
          const int r0 = k0 + t;       // K row for lanes 0-15
          const int r1 = k0 + t + 2;   // K row for lanes 16-31
          const float u0 = __shfl(a[r0 & 7], col + ((r0 >= 8) ? 16 : 0), 32);
          const float u1 = __shfl(a[r1 & 7], col + ((r1 >= 8) ? 16 : 0), 32);
          float uv = lh ? u1 : u0;
          if (col < kb) uv = 0.f;
          if (t == 0) bx = uv; else by = uv;
        }
        // f32 WMMA has no A-negate (NEG[1:0] must be 0) -> pre-negate A
        v2f Aop; Aop.x = -ax; Aop.y = -ay;
        v2f Bop; Bop.x = bx;  Bop.y = by;
        v8f Cop;
#pragma unroll
        for (int r = 0; r < 8; ++r) Cop[r] = a[r];
        Cop = __builtin_amdgcn_wmma_f32_16x16x4_f32(
            false, Aop, false, Bop, (short)0, Cop, false, false);
#pragma unroll
        for (int r = 0; r < 8; ++r) a[r] = Cop[r];
      }
    }

    if (lane == 0) out[(size_t)b * NCFG + c] = det;
  }
}

extern "C" void kernel_launch(void* const* d_in, const int* in_sizes, int n_in,
                              void* d_out, int out_size, void* d_ws, size_t ws_size,
                              hipStream_t stream) {
  const float* inp = (const float*)d_in[0];
  const int*   cfg = (const int*)d_in[1];
  float*       out = (float*)d_out;
  const int B = in_sizes[0] / (MDIM * MDIM);  // 1024
  slater_det_kernel<<<B, 256, 0, stream>>>(inp, cfg, out);
}